// ChemGCLayer_61907658604753
// MI455X (gfx1250) — compile-verified
//
#include <hip/hip_runtime.h>
#include <hip/hip_bf16.h>

// ---------------------------------------------------------------------------
// ChemGCLayer on gfx1250:  nfeats = X@W1+b1 ; gc = GCNConv(nfeats) ;
//                          out = [nfeats||gc] @ W2 + b2
// fp32 WMMA (v_wmma_f32_16x16x4_f32) for all three GEMMs, f32 global atomics
// for the scatter-add (hits the 192MB L2).
// ---------------------------------------------------------------------------

typedef __attribute__((ext_vector_type(2))) float v2f;
typedef __attribute__((ext_vector_type(8))) float v8f;

#define D_IN  128
#define D_HID 128
#define D_CAT 256
#define D_OUT 256

// ---------------------------------------------------------------------------
// GEMM: C[M x 128] = A[M x 128] @ B[128 x 128] (+ bias)
// 1 block = 16 rows of A staged in LDS; 8 waves, each computes one 16x16 tile.
// ---------------------------------------------------------------------------
__global__ __launch_bounds__(256)
void gemm_k128_n128(const float* __restrict__ A, const float* __restrict__ B,
                    const float* __restrict__ bias, float* __restrict__ C, int M)
{
    __shared__ float As[16 * 132];            // pad 128 -> 132 (bank-conflict free)
    const int row0 = blockIdx.x * 16;

    // cooperative, coalesced load of the 16x128 A tile
    for (int i = threadIdx.x; i < 16 * 128; i += 256) {
        int r = i >> 7, c = i & 127;
        int rr = row0 + r; if (rr >= M) rr = M - 1;
        As[r * 132 + c] = A[(size_t)rr * D_IN + c];
    }
    __syncthreads();

    const int wave = threadIdx.x >> 5;
    const int lane = threadIdx.x & 31;
    const int mn   = lane & 15;               // A row index / B col index in tile
    const int khalf = (lane >> 4) * 2;        // lanes 0-15 -> K+{0,1}, 16-31 -> K+{2,3}
    const int col0 = wave * 16;

    v8f acc = {};
    for (int k0 = 0; k0 < 128; k0 += 4) {
        v2f a, b;
        a.x = As[mn * 132 + k0 + khalf + 0];
        a.y = As[mn * 132 + k0 + khalf + 1];
        b.x = B[(size_t)(k0 + khalf + 0) * 128 + col0 + mn];
        b.y = B[(size_t)(k0 + khalf + 1) * 128 + col0 + mn];
        acc = __builtin_amdgcn_wmma_f32_16x16x4_f32(
                  false, a, false, b, (short)0, acc, false, false);
    }

    // C/D layout: VGPR r -> row (r + (lane<16 ? 0 : 8)), col = lane%16
    const int crow = (lane < 16) ? 0 : 8;
    const int ccol = col0 + mn;
    const float bv = bias ? bias[ccol] : 0.0f;
#pragma unroll
    for (int r = 0; r < 8; ++r) {
        int orow = row0 + crow + r;
        if (orow < M) C[(size_t)orow * 128 + ccol] = acc[r] + bv;
    }
}

// ---------------------------------------------------------------------------
// Final GEMM: out[M x 256] = concat(nf, gc+bg)[M x 256] @ W2[256 x 256] + b2
// 1 block = 16 rows; 8 waves, each computes two 16x16 tiles (32 cols).
// ---------------------------------------------------------------------------
__global__ __launch_bounds__(256)
void gemm_final(const float* __restrict__ nf, const float* __restrict__ gc,
                const float* __restrict__ bg, const float* __restrict__ W2,
                const float* __restrict__ b2, float* __restrict__ out, int M)
{
    __shared__ float As[16 * 260];            // pad 256 -> 260
    const int row0 = blockIdx.x * 16;

    for (int i = threadIdx.x; i < 16 * 256; i += 256) {
        int r = i >> 8, c = i & 255;
        int rr = row0 + r; if (rr >= M) rr = M - 1;
        float v;
        if (c < 128) v = nf[(size_t)rr * 128 + c];
        else         v = gc[(size_t)rr * 128 + (c - 128)] + bg[c - 128];
        As[r * 260 + c] = v;
    }
    __syncthreads();

    const int wave = threadIdx.x >> 5;
    const int lane = threadIdx.x & 31;
    const int mn   = lane & 15;
    const int khalf = (lane >> 4) * 2;
    const int col0 = wave * 32;

    v8f acc0 = {}, acc1 = {};
    for (int k0 = 0; k0 < 256; k0 += 4) {
        v2f a, b0, b1;
        a.x = As[mn * 260 + k0 + khalf + 0];
        a.y = As[mn * 260 + k0 + khalf + 1];
        const float* Br0 = W2 + (size_t)(k0 + khalf + 0) * 256;
        const float* Br1 = W2 + (size_t)(k0 + khalf + 1) * 256;
        b0.x = Br0[col0 + mn];       b0.y = Br1[col0 + mn];
        b1.x = Br0[col0 + 16 + mn];  b1.y = Br1[col0 + 16 + mn];
        acc0 = __builtin_amdgcn_wmma_f32_16x16x4_f32(
                   false, a, false, b0, (short)0, acc0, false, false);
        acc1 = __builtin_amdgcn_wmma_f32_16x16x4_f32(
                   false, a, false, b1, (short)0, acc1, false, false);
    }

    const int crow = (lane < 16) ? 0 : 8;
#pragma unroll
    for (int r = 0; r < 8; ++r) {
        int orow = row0 + crow + r;
        if (orow < M) {
            int c0 = col0 + mn, c1 = col0 + 16 + mn;
            out[(size_t)orow * 256 + c0] = acc0[r] + b2[c0];
            out[(size_t)orow * 256 + c1] = acc1[r] + b2[c1];
        }
    }
}

// ---------------------------------------------------------------------------
// Degree (edges only; self-loop folded in as +1 inside rsqrt)
// ---------------------------------------------------------------------------
__global__ void deg_kernel(const int* __restrict__ dst, float* __restrict__ deg, int E)
{
    int i = blockIdx.x * blockDim.x + threadIdx.x;
    if (i < E) atomicAdd(&deg[dst[i]], 1.0f);
}

__global__ void dinv_kernel(float* __restrict__ deg, int N)
{
    int i = blockIdx.x * blockDim.x + threadIdx.x;
    if (i < N) deg[i] = rsqrtf(deg[i] + 1.0f);   // +1 = self loop; deg>0 always
}

// ---------------------------------------------------------------------------
// Scatter: one wave per (edge | self-loop); lane handles 4 consecutive feats.
// acc[dst] += xg[src] * dinv[src]*dinv[dst]
// ---------------------------------------------------------------------------
__global__ __launch_bounds__(256)
void scatter_kernel(const int* __restrict__ src, const int* __restrict__ dst,
                    const float* __restrict__ dinv, const float* __restrict__ xg,
                    float* __restrict__ acc, int E, int N)
{
    int idx  = blockIdx.x * 8 + (threadIdx.x >> 5);
    int lane = threadIdx.x & 31;
    if (idx >= E + N) return;

    int s, d;
    if (idx < E) { s = src[idx]; d = dst[idx]; }
    else         { s = idx - E;  d = s; }

    float nrm = dinv[s] * dinv[d];
    const float4 v = ((const float4*)(xg + (size_t)s * 128))[lane];
    float* o = acc + (size_t)d * 128 + lane * 4;
    atomicAdd(o + 0, v.x * nrm);
    atomicAdd(o + 1, v.y * nrm);
    atomicAdd(o + 2, v.z * nrm);
    atomicAdd(o + 3, v.w * nrm);
}

// ---------------------------------------------------------------------------
extern "C" void kernel_launch(void* const* d_in, const int* in_sizes, int n_in,
                              void* d_out, int out_size, void* d_ws, size_t ws_size,
                              hipStream_t stream)
{
    // inputs (setup_inputs order): feats, edges, batch, W1, b1, Wg, bg, W2, b2
    const float* feats = (const float*)d_in[0];
    const int*   edges = (const int*)d_in[1];
    const int*   batch = (const int*)d_in[2];
    const float* W1 = (const float*)d_in[3];
    const float* b1 = (const float*)d_in[4];
    const float* Wg = (const float*)d_in[5];
    const float* bg = (const float*)d_in[6];
    const float* W2 = (const float*)d_in[7];
    const float* b2 = (const float*)d_in[8];

    const int M = in_sizes[0] / D_IN;      // 50000 nodes
    const int E = in_sizes[1] / 2;         // 800000 edges
    const int* e_src = edges;              // edges[0]
    const int* e_dst = edges + E;          // edges[1]

    // workspace layout
    char* ws = (char*)d_ws;
    size_t featBytes = (size_t)M * 128 * sizeof(float);
    float* nfeats = (float*)(ws);                      // M x 128
    float* xg     = (float*)(ws + featBytes);          // M x 128
    float* gcacc  = (float*)(ws + 2 * featBytes);      // M x 128
    float* deg    = (float*)(ws + 3 * featBytes);      // M

    // 1) zero the accumulators (must happen every call: atomics accumulate)
    hipMemsetAsync(gcacc, 0, featBytes, stream);
    hipMemsetAsync(deg, 0, (size_t)M * sizeof(float), stream);

    const int mtiles = (M + 15) / 16;

    // 2) nfeats = feats @ W1 + b1
    gemm_k128_n128<<<mtiles, 256, 0, stream>>>(feats, W1, b1, nfeats, M);

    // 3) xg = nfeats @ Wg  (bias bg applied after aggregation)
    gemm_k128_n128<<<mtiles, 256, 0, stream>>>(nfeats, Wg, nullptr, xg, M);

    // 4) degree + dinv
    deg_kernel<<<(E + 255) / 256, 256, 0, stream>>>(e_dst, deg, E);
    dinv_kernel<<<(M + 255) / 256, 256, 0, stream>>>(deg, M);

    // 5) scatter messages (edges + self loops)
    int nwork = E + M;
    scatter_kernel<<<(nwork + 7) / 8, 256, 0, stream>>>(e_src, e_dst, deg, xg,
                                                        gcacc, E, M);

    // 6) combined = [nfeats || gcacc + bg] @ W2 + b2  -> d_out
    float* combined = (float*)d_out;
    gemm_final<<<mtiles, 256, 0, stream>>>(nfeats, gcacc, bg, W2, b2, combined, M);

    // 7) pass-through outputs: edges, batch (raw copies after combined)
    char* outb = (char*)d_out;
    size_t combBytes = (size_t)M * D_OUT * sizeof(float);
    hipMemcpyAsync(outb + combBytes, d_in[1],
                   (size_t)in_sizes[1] * sizeof(int), hipMemcpyDeviceToDevice, stream);
    hipMemcpyAsync(outb + combBytes + (size_t)in_sizes[1] * sizeof(int), d_in[2],
                   (size_t)in_sizes[2] * sizeof(int), hipMemcpyDeviceToDevice, stream);
}